// FallbackImpl_52879637348697
// MI455X (gfx1250) — compile-verified
//
#include <hip/hip_runtime.h>

typedef __attribute__((ext_vector_type(2))) float v2f;
typedef __attribute__((ext_vector_type(8))) float v8f;

#define FDIM 576

// K-row -> (a index, b index) maps for the 56 pair-product rows of P.
// a[0]=seg0, a[1..3]=seg1, a[4..8]=seg2 (same for b).
__device__ static constexpr int IA[56] = {
    0,                                  // k0 : a0*b0
    1,1,1, 2,2,2, 3,3,3,                // k1-9 : a1[i]*b1[j]
    4,4,4,4,4, 5,5,5,5,5, 6,6,6,6,6, 7,7,7,7,7, 8,8,8,8,8, // k10-34: a2[i]*b2[j]
    0,0,0,                              // k35-37: a0*b1[j]
    1,2,3,                              // k38-40: a1[i]*b0
    4,4,4, 5,5,5, 6,6,6, 7,7,7, 8,8,8  // k41-55: a2[i]*b1[j]
};
__device__ static constexpr int IB[56] = {
    0,
    1,2,3, 1,2,3, 1,2,3,
    4,5,6,7,8, 4,5,6,7,8, 4,5,6,7,8, 4,5,6,7,8, 4,5,6,7,8,
    1,2,3,
    0,0,0,
    1,2,3, 1,2,3, 1,2,3, 1,2,3, 1,2,3
};

// Permuted index for the B-operand layout of V_WMMA_F32_16X16X4_F32:
// chunk kc holds K rows {4kc..4kc+3}; VGPR0 = rows {4kc, 4kc+2} (lane halves),
// VGPR1 = rows {4kc+1, 4kc+3}. Store so a lane loads its float2 contiguously:
// idx(k, col) = ((k>>2)*2 + ((k>>1)&1))*32 + col*2 + (k&1)
__global__ void build_w_kernel(const float* __restrict__ cg_a0,
                               const float* __restrict__ cg_a1,
                               const float* __restrict__ cg_a2,
                               const float* __restrict__ cg_a3,
                               const float* __restrict__ cg_b0,
                               const float* __restrict__ cg_b1,
                               const float* __restrict__ cg_b2,
                               float* __restrict__ W) {
  const int t = threadIdx.x;
  for (int i = t; i < 56 * 16; i += blockDim.x) W[i] = 0.0f;
  __syncthreads();

  int k = -1, col = 0;
  float v = 0.0f;
  if (t == 0)        { k = 0; col = 0; v = cg_a0[0]; }
  else if (t < 10)   { int u = t - 1;   k = 1 + u;        col = 0;          v = cg_a1[u]; }
  else if (t < 37)   { int u = t - 10;  k = 1 + u / 3;    col = 1 + u % 3;  v = cg_a2[u]; }
  else if (t < 162)  { int u = t - 37;  k = 10 + u / 5;   col = 4 + u % 5;  v = cg_a3[u]; }
  else if (t < 171)  { int u = t - 162; k = 35 + u / 3;   col = 1 + u % 3;  v = cg_b0[u]; }
  else if (t < 180)  { int u = t - 171; k = 38 + u / 3;   col = 1 + u % 3;  v = cg_b1[u]; }
  else if (t < 255)  { int u = t - 180; k = 41 + u / 5;   col = 4 + u % 5;  v = cg_b2[u]; }
  if (k >= 0) {
    int idx = ((k >> 2) * 2 + ((k >> 1) & 1)) * 32 + col * 2 + (k & 1);
    W[idx] = v;
  }
}

__global__ void __launch_bounds__(256)
tp_wmma_kernel(const float* __restrict__ x0, const int* __restrict__ i0,
               const float* __restrict__ x1, const float* __restrict__ W,
               float* __restrict__ out, int E) {
  const int lane = threadIdx.x & 31;
  const int wid  = threadIdx.x >> 5;
  // One wave per edge; wave-uniform so EXEC stays all-ones for WMMA.
  const int e = __builtin_amdgcn_readfirstlane((int)(blockIdx.x * 8 + wid));
  if (e >= E) return;

  const int  col = lane & 15;
  const bool hi  = lane >= 16;

  // Load B operands (shared CG weight matrix) once per wave: 14 x float2.
  v2f wv[14];
  {
    const float* wp = W + (hi ? 32 : 0) + col * 2;
#pragma unroll
    for (int kc = 0; kc < 14; ++kc)
      wv[kc] = *(const v2f*)(wp + kc * 64);
  }

  const int node = i0[e];  // uniform address -> scalar load
  const float* __restrict__ arow = x0 + (size_t)node * FDIM;
  const float* __restrict__ brow = x1 + (size_t)e * FDIM;
  float* __restrict__ orow       = out + (size_t)e * FDIM;

  // Output column -> feature-offset mapping (cols 0..8 valid).
  int obase, od, oj;
  if (col == 0)     { obase = 0;   od = 1; oj = 0; }
  else if (col < 4) { obase = 64;  od = 3; oj = col - 1; }
  else              { obase = 256; od = 5; oj = col - 4; }
  const bool active = col < 9;

  for (int q = 0; q < 4; ++q) {
    const int m = q * 16 + col;  // lane pair (L, L+16) handles item row L
    float a[9], b[9];
    a[0] = arow[m];
    b[0] = brow[m];
#pragma unroll
    for (int j = 0; j < 3; ++j) {
      a[1 + j] = arow[64 + 3 * m + j];
      b[1 + j] = brow[64 + 3 * m + j];
    }
#pragma unroll
    for (int j = 0; j < 5; ++j) {
      a[4 + j] = arow[256 + 5 * m + j];
      b[4 + j] = brow[256 + 5 * m + j];
    }

    v8f c = {0.f, 0.f, 0.f, 0.f, 0.f, 0.f, 0.f, 0.f};
#pragma unroll
    for (int kc = 0; kc < 14; ++kc) {
      const int k0 = 4 * kc;
      // A-matrix 16x4 layout: lanes 0-15 hold K = {4kc, 4kc+1},
      // lanes 16-31 hold K = {4kc+2, 4kc+3} for row = lane&15.
      float plo0 = a[IA[k0 + 0]] * b[IB[k0 + 0]];
      float plo1 = a[IA[k0 + 1]] * b[IB[k0 + 1]];
      float phi0 = a[IA[k0 + 2]] * b[IB[k0 + 2]];
      float phi1 = a[IA[k0 + 3]] * b[IB[k0 + 3]];
      v2f av;
      av.x = hi ? phi0 : plo0;
      av.y = hi ? phi1 : plo1;
      c = __builtin_amdgcn_wmma_f32_16x16x4_f32(
          /*neg_a=*/false, av, /*neg_b=*/false, wv[kc],
          /*c_mod=*/(short)0, c, /*reuse_a=*/false, /*reuse_b=*/false);
    }

    if (active) {
      // D layout: VGPR v = row v (lanes 0-15) / row v+8 (lanes 16-31), N = col.
#pragma unroll
      for (int v = 0; v < 8; ++v) {
        const int mr = q * 16 + v + (hi ? 8 : 0);
        orow[obase + od * mr + oj] = c[v];
      }
    }
  }
}

extern "C" void kernel_launch(void* const* d_in, const int* in_sizes, int n_in,
                              void* d_out, int out_size, void* d_ws, size_t ws_size,
                              hipStream_t stream) {
  const float* x0 = (const float*)d_in[0];
  const int*   i0 = (const int*)d_in[1];
  const float* x1 = (const float*)d_in[2];
  const float* cg_a0 = (const float*)d_in[3];
  const float* cg_a1 = (const float*)d_in[4];
  const float* cg_a2 = (const float*)d_in[5];
  const float* cg_a3 = (const float*)d_in[6];
  const float* cg_b0 = (const float*)d_in[7];
  const float* cg_b1 = (const float*)d_in[8];
  const float* cg_b2 = (const float*)d_in[9];

  const int E = in_sizes[1];
  float* W = (float*)d_ws;  // 56*16 floats = 3584 bytes

  build_w_kernel<<<1, 256, 0, stream>>>(cg_a0, cg_a1, cg_a2, cg_a3,
                                        cg_b0, cg_b1, cg_b2, W);

  const int blocks = (E + 7) / 8;  // 8 waves (edges) per 256-thread block
  tp_wmma_kernel<<<blocks, 256, 0, stream>>>(x0, i0, x1, W, (float*)d_out, E);
}